// LocalPatchMixer_6794638262344
// MI455X (gfx1250) — compile-verified
//
#include <hip/hip_runtime.h>

#ifndef __has_builtin
#define __has_builtin(x) 0
#endif

#if defined(__HIP_DEVICE_COMPILE__) && \
    __has_builtin(__builtin_amdgcn_tensor_load_to_lds) && \
    __has_builtin(__builtin_amdgcn_s_wait_tensorcnt)
#define USE_TDM 1
#else
#define USE_TDM 0
#endif

namespace {
constexpr int kB = 4;
constexpr int kN = 16384;
constexpr int kD = 256;
constexpr int kG = 256;                 // GRID
constexpr int kCells = kG * kG;         // 65536
constexpr int kPPB = 8;                 // points per block in fused kernel
constexpr int kWB = kD * 9;             // weight float count = 2304
constexpr int kSMemFloats = kWB + kD;   // weights + bias = 2560 floats
}  // namespace

// ---------------------------------------------------------------------------
// Workspace layout (ints):  owner[kB*kCells]  |  mins[2*kB]  (minx,miny / batch)
// ---------------------------------------------------------------------------

__global__ __launch_bounds__(256) void lpm_init(int* __restrict__ owner,
                                                int* __restrict__ mins) {
  int i = blockIdx.x * blockDim.x + threadIdx.x;
  if (i < kB * kCells) owner[i] = -1;
  if (i < 2 * kB) mins[i] = 0x7fffffff;
}

__global__ __launch_bounds__(256) void lpm_minred(const int2* __restrict__ coords,
                                                  int* __restrict__ mins) {
  __shared__ int sx[256];
  __shared__ int sy[256];
  const int tid = threadIdx.x;
  const int pid = blockIdx.x * 256 + tid;      // kN % 256 == 0 -> block is batch-pure
  int2 c = coords[pid];
  sx[tid] = c.x;
  sy[tid] = c.y;
  __syncthreads();
#pragma unroll
  for (int s = 128; s > 0; s >>= 1) {
    if (tid < s) {
      sx[tid] = min(sx[tid], sx[tid + s]);
      sy[tid] = min(sy[tid], sy[tid + s]);
    }
    __syncthreads();
  }
  if (tid == 0) {
    int b = pid / kN;
    atomicMin(&mins[2 * b + 0], sx[0]);
    atomicMin(&mins[2 * b + 1], sy[0]);
  }
}

__global__ __launch_bounds__(256) void lpm_owner(const int2* __restrict__ coords,
                                                 const int* __restrict__ mins,
                                                 int* __restrict__ owner) {
  const int pid = blockIdx.x * blockDim.x + threadIdx.x;
  if (pid >= kB * kN) return;
  const int b = pid / kN;
  const int n = pid - b * kN;
  int2 c = coords[pid];
  const int gx = c.x - mins[2 * b + 0];
  const int gy = c.y - mins[2 * b + 1];
  // last-write-wins on duplicates == largest point index wins
  atomicMax(&owner[b * kCells + gy * kG + gx], n);
}

#if USE_TDM
typedef unsigned int u32x4 __attribute__((ext_vector_type(4)));
typedef int i32x4 __attribute__((ext_vector_type(4)));
typedef int i32x8 __attribute__((ext_vector_type(8)));

// 1-D TDM copy: nElems f32 from global -> LDS byte offset ldsByteOff.
// D# built per CDNA5 ISA ch.8: group0 {count=1, lds_addr, global_addr, type=2},
// group1 {data_size=4B, tensor_dim0=nElems, tensor_dim1=1, tile_dim0=nElems,
//         tile_dim1=1, tensor_dim0_stride=nElems}; higher groups zero (1-D tile).
__device__ __forceinline__ void tdm_copy_row(const void* gptr,
                                             unsigned ldsByteOff,
                                             unsigned nElems) {
  unsigned long long ga = (unsigned long long)gptr;
  u32x4 g0;
  g0[0] = 1u;                                                   // count=1 (valid), user mode
  g0[1] = ldsByteOff;                                           // lds_addr (bytes)
  g0[2] = (unsigned)ga;                                         // global_addr[31:0]
  g0[3] = (unsigned)((ga >> 32) & 0x1FFFFFFull) | (2u << 30);   // global_addr[56:32] | type=2
  i32x8 g1;
  g1[0] = 0x20000;                                              // data_size=2 -> 4 bytes
  g1[1] = (int)((nElems & 0xFFFFu) << 16);                      // tensor_dim0[15:0] @ bits[63:48]
  g1[2] = (int)(((nElems >> 16) & 0xFFFFu) | (1u << 16));       // tensor_dim0[31:16] | tensor_dim1=1
  g1[3] = (int)((nElems & 0xFFFFu) << 16);                      // tile_dim0 @ bits[127:112]
  g1[4] = 1;                                                    // tile_dim1 = 1
  g1[5] = (int)nElems;                                          // tensor_dim0_stride[31:0]
  g1[6] = 0;
  g1[7] = 0;
  i32x4 z4 = {0, 0, 0, 0};
  i32x8 z8 = {0, 0, 0, 0, 0, 0, 0, 0};
  // 6-arg (clang-23 / therock) form: groups 2,3 + extra group zero-filled.
  __builtin_amdgcn_tensor_load_to_lds(g0, g1, z4, z4, z8, 0);
}
#endif

// Fused: depthwise 3x3 conv (via owner-index indirection) + bias + exact GELU
// + gather at point locations + residual.  One channel per lane, kPPB points/block.
__global__ __launch_bounds__(256) void lpm_conv_gather(
    const float* __restrict__ x, const int2* __restrict__ coords,
    const float* __restrict__ conv_w, const float* __restrict__ conv_b,
    const int* __restrict__ mins, const int* __restrict__ owner,
    float* __restrict__ out) {
  __shared__ __align__(16) float smem[kSMemFloats + 16];
  float* s_w = smem;                        // [kD*9], stride-9 -> LDS conflict-free
  float* s_b = smem + kWB;                  // [kD]
  int* s_tap = (int*)(smem + kSMemFloats);  // [9] owner indices of the 3x3 taps

  const int lane = threadIdx.x;  // channel 0..255

#if USE_TDM
  if (lane == 0) {
    tdm_copy_row(conv_w, 0u, (unsigned)kWB);                 // weights to LDS bytes 0..9215
    tdm_copy_row(conv_b, (unsigned)(kWB * 4), (unsigned)kD); // bias to LDS from byte 9216
  }
  __builtin_amdgcn_s_wait_tensorcnt(0);
  __syncthreads();
#else
  for (int i = lane; i < kSMemFloats; i += 256)
    smem[i] = (i < kWB) ? conv_w[i] : conv_b[i - kWB];
  __syncthreads();
#endif

  const int pid0 = blockIdx.x * kPPB;
  for (int p = 0; p < kPPB; ++p) {
    const int pid = pid0 + p;
    const int b = pid / kN;
    const int n = pid - b * kN;
    const size_t xbase = (size_t)b * kN * kD;
    const size_t ri = xbase + (size_t)n * kD + lane;

    __syncthreads();  // protect s_tap reuse across points
    __builtin_prefetch(&x[ri], 0, 3);  // global_prefetch_b8 of residual row
    if (lane < 9) {
      int2 c = coords[pid];
      // cross-correlation taps: out[y,x] += w[ky,kx] * in[y+ky-1, x+kx-1]
      int yy = c.y - mins[2 * b + 1] + lane / 3 - 1;
      int xx = c.x - mins[2 * b + 0] + lane % 3 - 1;
      int o = -1;
      if ((unsigned)yy < (unsigned)kG && (unsigned)xx < (unsigned)kG)
        o = owner[b * kCells + yy * kG + xx];
      s_tap[lane] = o;  // -1 == empty cell == zero background
    }
    __syncthreads();

    float acc = 0.0f;
#pragma unroll
    for (int t = 0; t < 9; ++t) {
      int o = s_tap[t];  // block-uniform -> wave-uniform branch
      if (o >= 0)
        acc = fmaf(s_w[lane * 9 + t], x[xbase + (size_t)o * kD + lane], acc);
    }
    float v = acc + s_b[lane];
    float g = 0.5f * v * (1.0f + erff(v * 0.70710678118654752f));  // exact GELU
    out[ri] = x[ri] + g;
  }
}

extern "C" void kernel_launch(void* const* d_in, const int* in_sizes, int n_in,
                              void* d_out, int out_size, void* d_ws, size_t ws_size,
                              hipStream_t stream) {
  const float* x      = (const float*)d_in[0];
  const int2*  coords = (const int2*)d_in[1];   // (B,N,2) int32 -> int2 {gx,gy}
  const float* conv_w = (const float*)d_in[2];  // (D,1,3,3)
  const float* conv_b = (const float*)d_in[3];  // (D,)
  float* out = (float*)d_out;

  int* owner = (int*)d_ws;            // kB*kCells ints (~1 MiB)
  int* mins  = owner + kB * kCells;   // 2*kB ints

  (void)in_sizes; (void)n_in; (void)out_size; (void)ws_size;

  lpm_init<<<(kB * kCells + 255) / 256, 256, 0, stream>>>(owner, mins);
  lpm_minred<<<(kB * kN) / 256, 256, 0, stream>>>(coords, mins);
  lpm_owner<<<(kB * kN + 255) / 256, 256, 0, stream>>>(coords, mins, owner);
  lpm_conv_gather<<<(kB * kN) / kPPB, 256, 0, stream>>>(
      x, coords, conv_w, conv_b, mins, owner, out);
}